// CombinedRiemannDownstreamHead_79182017069220
// MI455X (gfx1250) — compile-verified
//
#include <hip/hip_runtime.h>
#include <hip/hip_bf16.h>

typedef __attribute__((ext_vector_type(2))) float v2f;
typedef __attribute__((ext_vector_type(8))) float v8f;

#define B_    32
#define L_    2048
#define D_    512
#define C_    64
#define N_    32
#define TRI_  2080
#define K_TOT 16384
#define COMB  2592

// workspace layout (floats)
#define OFF_POOLED 0
#define OFF_COV    (B_*D_)                 // 16384
#define OFF_TAN    (OFF_COV + B_*C_*C_)    // 147456

__global__ void k_zero(float* ws, int n) {
    int i = blockIdx.x * blockDim.x + threadIdx.x;
    if (i < n) ws[i] = 0.0f;
}

// pooled[b,d] = (1/L) * sum_l LN(x[b,l,:])[d] * gamma + beta
// one wave per 16 rows; wave32 shuffle reduction for mean/var
__global__ __launch_bounds__(256) void k_pool_ln(const float* __restrict__ x,
                                                 const float* __restrict__ gamma,
                                                 const float* __restrict__ beta,
                                                 float* __restrict__ pooled) {
    const int b    = blockIdx.x >> 4;      // 16 blocks per batch
    const int blk  = blockIdx.x & 15;
    const int wave = threadIdx.x >> 5;
    const int lane = threadIdx.x & 31;

    float g[16], be[16], acc[16];
#pragma unroll
    for (int t = 0; t < 16; ++t) {
        g[t]   = gamma[lane + 32*t];
        be[t]  = beta [lane + 32*t];
        acc[t] = 0.0f;
    }
    const int wid = blk * 8 + wave;        // 0..127 per batch
    const float* xb = x + (size_t)b * L_ * D_;

    for (int r = 0; r < 16; ++r) {
        const int l = wid * 16 + r;
        const float* xr = xb + (size_t)l * D_;
        float v[16], s = 0.0f, ss = 0.0f;
#pragma unroll
        for (int t = 0; t < 16; ++t) {
            v[t] = xr[lane + 32*t];
            s  += v[t];
            ss += v[t] * v[t];
        }
#pragma unroll
        for (int m = 16; m >= 1; m >>= 1) {
            s  += __shfl_xor(s,  m, 32);
            ss += __shfl_xor(ss, m, 32);
        }
        const float mu  = s  * (1.0f / (float)D_);
        const float var = ss * (1.0f / (float)D_) - mu * mu;
        const float rs  = rsqrtf(var + 1e-5f);
#pragma unroll
        for (int t = 0; t < 16; ++t)
            acc[t] += (v[t] - mu) * rs * g[t] + be[t];
    }
#pragma unroll
    for (int t = 0; t < 16; ++t)
        atomicAdd(&pooled[b * D_ + lane + 32*t], acc[t] * (1.0f / (float)L_));
}

// Full-Gram SYRK per wave: one wave computes the entire 64x64 tile for one
// (batch, n-panel, d-half). SYRK symmetry => A-frag and B-frag load patterns
// are identical, so each K-step needs only 4 global_load_b64 per lane feeding
// 16 V_WMMA_F32_16X16X4_F32 (0.25 loads/WMMA vs 2 in the naive version).
// A frag (16x4 f32 layout): lanes 0-15 hold M=0..15 K=0,1; lanes 16-31 K=2,3.
__global__ __launch_bounds__(32) void k_cov_wmma(const float* __restrict__ x,
                                                 float* __restrict__ cov) {
    const int dsplit = blockIdx.x;         // 0..1   (d-halves of the panel)
    const int n      = blockIdx.y;         // 0..31  (n-panels)
    const int b      = blockIdx.z;         // 0..31
    const int lane   = threadIdx.x;
    const int rsel   = lane & 15;
    const int koff   = (lane >> 4) * 2;

    const float* panel = x + ((size_t)b * L_ + (size_t)n * C_) * D_;  // 64 x 512

    v8f acc[4][4];
#pragma unroll
    for (int i = 0; i < 4; ++i)
#pragma unroll
        for (int j = 0; j < 4; ++j) {
            v8f z = {};
            acc[i][j] = z;
        }

    const float* base[4];
#pragma unroll
    for (int t = 0; t < 4; ++t)
        base[t] = panel + (size_t)(t * 16 + rsel) * D_ + koff;

    const int d0 = dsplit * (D_ / 2);
    for (int d = d0; d < d0 + D_ / 2; d += 4) {
        v2f f[4];
#pragma unroll
        for (int t = 0; t < 4; ++t)
            f[t] = *reinterpret_cast<const v2f*>(base[t] + d);
#pragma unroll
        for (int i = 0; i < 4; ++i)
#pragma unroll
            for (int j = 0; j < 4; ++j)
                acc[i][j] = __builtin_amdgcn_wmma_f32_16x16x4_f32(
                    false, f[i], false, f[j], (short)0, acc[i][j], false, false);
    }

    const float sc = 1.0f / (float)K_TOT;
    float* cb = cov + (size_t)b * C_ * C_;
#pragma unroll
    for (int i = 0; i < 4; ++i)
#pragma unroll
        for (int j = 0; j < 4; ++j) {
#pragma unroll
            for (int v = 0; v < 8; ++v) {
                const int row = i * 16 + v + ((lane >> 4) << 3);  // C/D layout
                const int col = j * 16 + rsel;
                atomicAdd(&cb[row * C_ + col], acc[i][j][v] * sc);
            }
        }
}

// Per batch: Gauss-Jordan solve (cov+eI+I) X = 2(cov+eI-I) in LDS,
// symmetrize, upper-triangle, LayerNorm -> tangent
__global__ __launch_bounds__(256) void k_cayley_tangent(const float* __restrict__ cov,
                                                        const float* __restrict__ gamma_t,
                                                        const float* __restrict__ beta_t,
                                                        float* __restrict__ tangent) {
    __shared__ float M [C_][C_ + 1];
    __shared__ float Rm[C_][C_ + 1];
    __shared__ float fcol[C_];
    __shared__ float tanv[TRI_];
    __shared__ float red1[256];
    __shared__ float red2[256];

    const int b   = blockIdx.x;
    const int tid = threadIdx.x;
    const float* cb = cov + (size_t)b * C_ * C_;

    for (int p = tid; p < C_ * C_; p += 256) {
        int i = p >> 6, j = p & 63;
        float cv = cb[p] + ((i == j) ? 1e-5f : 0.0f);   // ridge
        M [i][j] = cv + ((i == j) ? 1.0f : 0.0f);
        Rm[i][j] = 2.0f * (cv - ((i == j) ? 1.0f : 0.0f));
    }
    __syncthreads();

    for (int k = 0; k < C_; ++k) {
        float inv = 1.0f / M[k][k];
        float fac = 0.0f;
        if (tid < C_ && tid != k) fac = M[tid][k];
        __syncthreads();
        if (tid < C_) { M[k][tid] *= inv; fcol[tid] = fac; }
        else if (tid < 2 * C_) Rm[k][tid - C_] *= inv;
        __syncthreads();
        const int r = tid >> 2, q = tid & 3;
        if (r != k) {
            float fr = fcol[r];
            for (int c = q; c < C_; c += 4) {
                M [r][c] -= fr * M [k][c];
                Rm[r][c] -= fr * Rm[k][c];
            }
        }
        __syncthreads();
    }

    // upper triangle of 0.5*(Lm + Lm^T) + stats
    float s = 0.0f, ss = 0.0f;
    for (int p = tid; p < TRI_; p += 256) {
        int i = 0, base = 0;
        while (p - base >= C_ - i) { base += C_ - i; ++i; }
        int j = i + (p - base);
        float v = 0.5f * (Rm[i][j] + Rm[j][i]);
        tanv[p] = v;
        s += v; ss += v * v;
    }
    red1[tid] = s; red2[tid] = ss;
    __syncthreads();
    for (int off = 128; off >= 1; off >>= 1) {
        if (tid < off) { red1[tid] += red1[tid + off]; red2[tid] += red2[tid + off]; }
        __syncthreads();
    }
    const float mu  = red1[0] * (1.0f / (float)TRI_);
    const float var = red2[0] * (1.0f / (float)TRI_) - mu * mu;
    const float rs  = rsqrtf(var + 1e-5f);
    for (int p = tid; p < TRI_; p += 256)
        tangent[(size_t)b * TRI_ + p] = (tanv[p] - mu) * rs * gamma_t[p] + beta_t[p];
}

// out[b,:] = [pooled | tangent] @ W + bias
__global__ __launch_bounds__(256) void k_head(const float* __restrict__ pooled,
                                              const float* __restrict__ tangent,
                                              const float* __restrict__ W,
                                              const float* __restrict__ bias,
                                              float* __restrict__ out) {
    __shared__ float red[256 * 4];
    const int b = blockIdx.x, tid = threadIdx.x;
    float acc[4] = {0.0f, 0.0f, 0.0f, 0.0f};
    for (int k = tid; k < COMB; k += 256) {
        float c = (k < D_) ? pooled[b * D_ + k]
                           : tangent[(size_t)b * TRI_ + (k - D_)];
#pragma unroll
        for (int cls = 0; cls < 4; ++cls) acc[cls] += c * W[k * 4 + cls];
    }
#pragma unroll
    for (int cls = 0; cls < 4; ++cls) red[tid * 4 + cls] = acc[cls];
    __syncthreads();
    for (int off = 128; off >= 1; off >>= 1) {
        if (tid < off) {
#pragma unroll
            for (int cls = 0; cls < 4; ++cls)
                red[tid * 4 + cls] += red[(tid + off) * 4 + cls];
        }
        __syncthreads();
    }
    if (tid < 4) out[b * 4 + tid] = red[tid] + bias[tid];
}

extern "C" void kernel_launch(void* const* d_in, const int* in_sizes, int n_in,
                              void* d_out, int out_size, void* d_ws, size_t ws_size,
                              hipStream_t stream) {
    (void)in_sizes; (void)n_in; (void)out_size; (void)ws_size;
    const float* x      = (const float*)d_in[0];
    const float* g_pool = (const float*)d_in[1];
    const float* b_pool = (const float*)d_in[2];
    const float* g_tan  = (const float*)d_in[3];
    const float* b_tan  = (const float*)d_in[4];
    const float* W      = (const float*)d_in[5];
    const float* bias   = (const float*)d_in[6];

    float* ws      = (float*)d_ws;
    float* pooled  = ws + OFF_POOLED;
    float* cov     = ws + OFF_COV;
    float* tangent = ws + OFF_TAN;
    float* out     = (float*)d_out;

    const int nzero = OFF_TAN;   // pooled + cov accumulators
    k_zero<<<(nzero + 255) / 256, 256, 0, stream>>>(ws, nzero);

    k_pool_ln<<<B_ * 16, 256, 0, stream>>>(x, g_pool, b_pool, pooled);
    k_cov_wmma<<<dim3(2, N_, B_), 32, 0, stream>>>(x, cov);
    k_cayley_tangent<<<B_, 256, 0, stream>>>(cov, g_tan, b_tan, tangent);
    k_head<<<B_, 256, 0, stream>>>(pooled, tangent, W, bias, out);
}